// BinaryClassifier_2783138808289
// MI455X (gfx1250) — compile-verified
//
#include <hip/hip_runtime.h>

typedef __attribute__((ext_vector_type(2))) float v2f;
typedef __attribute__((ext_vector_type(8))) float v8f;

#define VSIZE 100000
#define DDIM  100
#define LTOK  200
#define BATCH 16384

// ---------------------------------------------------------------------------
// Pass 1: s[v] = sum_d T[v,d] * w[d], 16 rows per wave via V_WMMA_F32_16X16X4_F32.
// A (16x4 f32): lanes 0-15 hold (M=lane, K={k0,k0+1}), lanes 16-31 hold
//               (M=lane-16, K={k0+2,k0+3})  [ISA 7.12.2 32-bit A layout]
// B (4x16 f32): lane n<16 holds (K={k0,k0+1}, N=n); lane 16+n holds
//               (K={k0+2,k0+3}, N=n). Only column N=0 carries w[k].
// D (16x16 f32, 8 VGPRs): lane 0 holds (M=0..7, N=0); lane 16 holds (M=8..15, N=0).
// ---------------------------------------------------------------------------
__global__ __launch_bounds__(256) void rowdot_wmma(const float* __restrict__ T,
                                                   const float* __restrict__ w,
                                                   float* __restrict__ s) {
    const int lane  = threadIdx.x & 31;
    const int wave  = threadIdx.x >> 5;
    const int tile  = blockIdx.x * 8 + wave;      // 16 vocab rows per tile
    if (tile * 16 >= VSIZE) return;               // wave-uniform branch (EXEC stays full)

    const int m      = lane & 15;                 // row within tile / N within B
    const int khalf  = lane >> 4;                 // 0: K={0,1}; 1: K={2,3}
    const bool col0  = (m == 0);                  // lanes 0 and 16 hold B column N=0

    // Row base is (tile*16+m)*100 floats -> even float index -> 8B aligned.
    const v2f* __restrict__ rowp2 =
        (const v2f*)(T + (size_t)(tile * 16 + m) * DDIM);
    const v2f* __restrict__ w2 = (const v2f*)w;   // ka even -> pair index ka/2

    v8f acc = {};
    #pragma unroll 5
    for (int k0 = 0; k0 < DDIM; k0 += 4) {        // 25 k-steps cover D=100 exactly
        const int kp = (k0 >> 1) + khalf;         // pair index: (k0 + 2*khalf)/2
        v2f a  = rowp2[kp];                       // one global_load_b64
        v2f wv = w2[kp];                          // unconditional, in-bounds for all lanes
        v2f b;
        b.x = col0 ? wv.x : 0.0f;                 // v_cndmask, no divergent branch
        b.y = col0 ? wv.y : 0.0f;
        acc = __builtin_amdgcn_wmma_f32_16x16x4_f32(
            /*neg_a=*/false, a, /*neg_b=*/false, b,
            /*c_mod=*/(short)0, acc, /*reuse_a=*/false, /*reuse_b=*/false);
    }

    // Column N=0 of D: lane 0 -> rows 0..7, lane 16 -> rows 8..15.
    if (col0) {
        float* __restrict__ sp = s + tile * 16 + (khalf ? 8 : 0);
        #pragma unroll
        for (int i = 0; i < 8; ++i) sp[i] = acc[i];
    }
}

// ---------------------------------------------------------------------------
// Pass 2: out[b] = (1/L) * sum_l s[idx[b,l]].  One wave32 per batch row:
// coalesced index loads, random 4B gathers hit the 400KB s-table in L2.
// ---------------------------------------------------------------------------
__global__ __launch_bounds__(256) void pool_gather(const int* __restrict__ idx,
                                                   const float* __restrict__ s,
                                                   float* __restrict__ out) {
    const int lane = threadIdx.x & 31;
    const int wave = threadIdx.x >> 5;
    const int b    = blockIdx.x * 8 + wave;
    if (b >= BATCH) return;

    const int* __restrict__ ip = idx + (size_t)b * LTOK;
    float sum = 0.0f;
    for (int l = lane; l < LTOK; l += 32)
        sum += s[ip[l]];

    // wave32 butterfly reduction
    sum += __shfl_xor(sum, 16, 32);
    sum += __shfl_xor(sum,  8, 32);
    sum += __shfl_xor(sum,  4, 32);
    sum += __shfl_xor(sum,  2, 32);
    sum += __shfl_xor(sum,  1, 32);

    if (lane == 0) out[b] = sum * (1.0f / (float)LTOK);
}

// ---------------------------------------------------------------------------
// Fallback if workspace is too small for the 400KB s-table: naive fused path.
// ---------------------------------------------------------------------------
__global__ __launch_bounds__(256) void fused_naive(const int* __restrict__ idx,
                                                   const float* __restrict__ T,
                                                   const float* __restrict__ w,
                                                   float* __restrict__ out) {
    const int lane = threadIdx.x & 31;
    const int wave = threadIdx.x >> 5;
    const int b    = blockIdx.x * 8 + wave;
    if (b >= BATCH) return;

    const int* __restrict__ ip = idx + (size_t)b * LTOK;
    float sum = 0.0f;
    for (int l = 0; l < LTOK; ++l) {
        const float* __restrict__ rowp = T + (size_t)ip[l] * DDIM;
        for (int d = lane; d < DDIM; d += 32)
            sum += rowp[d] * w[d];
    }
    sum += __shfl_xor(sum, 16, 32);
    sum += __shfl_xor(sum,  8, 32);
    sum += __shfl_xor(sum,  4, 32);
    sum += __shfl_xor(sum,  2, 32);
    sum += __shfl_xor(sum,  1, 32);
    if (lane == 0) out[b] = sum * (1.0f / (float)LTOK);
}

extern "C" void kernel_launch(void* const* d_in, const int* in_sizes, int n_in,
                              void* d_out, int out_size, void* d_ws, size_t ws_size,
                              hipStream_t stream) {
    const int*   word_idxs = (const int*)d_in[0];    // [B, L] int32
    const float* embed     = (const float*)d_in[1];  // [V, D] fp32
    const float* weights   = (const float*)d_in[2];  // [D, 1] fp32
    float*       out       = (float*)d_out;          // [B, 1] fp32

    if (ws_size >= (size_t)VSIZE * sizeof(float)) {
        float* s = (float*)d_ws;                     // 400 KB row-dot table
        const int tiles   = (VSIZE + 15) / 16;       // 6250
        const int blocks1 = (tiles + 7) / 8;         // 8 waves/block
        rowdot_wmma<<<blocks1, 256, 0, stream>>>(embed, weights, s);

        const int blocks2 = (BATCH + 7) / 8;         // 2048
        pool_gather<<<blocks2, 256, 0, stream>>>(word_idxs, s, out);
    } else {
        const int blocks = (BATCH + 7) / 8;
        fused_naive<<<blocks, 256, 0, stream>>>(word_idxs, embed, weights, out);
    }
}